// Mix_Cholesky_13185549599171
// MI455X (gfx1250) — compile-verified
//
#include <hip/hip_runtime.h>
#include <hip/hip_bf16.h>
#include <math.h>

// ---------------------------------------------------------------------------
// Mix_Cholesky splat raster for MI455X (gfx1250, wave32).
//   sigma(pixel, gauss) via V_WMMA_F32_16X16X4_F32 : rank-3 factorization
//   log2w = [x^2, x, 1] . (-log2e)*[A, e1, e0]   (row-local coords, y'==0)
//   so v_exp_f32 consumes the WMMA result directly.
// ---------------------------------------------------------------------------

#define HH 256
#define WW 256
#define GG 4
#define NEG_LOG2E (-1.4426950408889634f)
#define INV_2PI   (0.15915494309189535f)

typedef float v2f __attribute__((ext_vector_type(2)));
typedef float v8f __attribute__((ext_vector_type(8)));

// -------------------- kernel 1: projection into padded AoS ws --------------
// base[slot][16] = {cx, cy, A(=.5*conic0), B(=conic1), C(=.5*conic2), opac, r, g, b}
// gab [hslot][16] = {fx0..fx3, fy0..fy3, w0..w3}   (fx,fy pre-scaled by 1/2pi)
__global__ __launch_bounds__(256) void project_kernel(
    const float* __restrict__ mu_l, const float* __restrict__ mu_h,
    const float* __restrict__ ch_l, const float* __restrict__ ch_h,
    const float* __restrict__ ft_l, const float* __restrict__ ft_h,
    const float* __restrict__ op_l, const float* __restrict__ op_h,
    const float* __restrict__ gf,   const float* __restrict__ gw,
    float* __restrict__ base, float* __restrict__ gab,
    int nl, int nh, int NLpad, int NHpad)
{
    int i = blockIdx.x * blockDim.x + threadIdx.x;
    int tot = NLpad + NHpad;
    if (i >= tot) return;

    bool isHigh = (i >= NLpad);
    int idx = isHigh ? (i - NLpad) : i;
    int n   = isHigh ? nh : nl;

    float o[9];
    #pragma unroll
    for (int j = 0; j < 9; ++j) o[j] = 0.0f;

    if (idx < n) {
        const float* mu = isHigh ? mu_h : mu_l;
        const float* ch = isHigh ? ch_h : ch_l;
        const float* ft = isHigh ? ft_h : ft_l;
        const float* op = isHigh ? op_h : op_l;

        float m0 = tanhf(mu[idx * 2 + 0]);
        float m1 = tanhf(mu[idx * 2 + 1]);
        o[0] = (m0 + 1.0f) * 0.5f * (float)WW;   // cx
        o[1] = (m1 + 1.0f) * 0.5f * (float)HH;   // cy

        float l1 = ch[idx * 3 + 0] + 0.5f;
        float l2 = ch[idx * 3 + 1] + 0.0f;
        float l3 = ch[idx * 3 + 2] + 0.5f;
        float sxx = l1 * l1;
        float sxy = l1 * l2;
        float syy = l2 * l2 + l3 * l3;
        float det = sxx * syy - sxy * sxy;
        float inv = 1.0f / det;
        o[2] = 0.5f * (syy * inv);               // A = 0.5*conic0
        o[3] = -(sxy * inv);                     // B = conic1
        o[4] = 0.5f * (sxx * inv);               // C = 0.5*conic2
        o[5] = op[idx];                          // opac (0 padding -> contrib 0)
        o[6] = ft[idx * 3 + 0];
        o[7] = ft[idx * 3 + 1];
        o[8] = ft[idx * 3 + 2];
    }

    float* bp = base + (size_t)i * 16;
    #pragma unroll
    for (int j = 0; j < 9; ++j) bp[j] = o[j];

    if (isHigh) {
        float g12[12];
        #pragma unroll
        for (int j = 0; j < 12; ++j) g12[j] = 0.0f;
        if (idx < nh) {
            #pragma unroll
            for (int j = 0; j < GG; ++j) {
                g12[j]     = gf[(idx * GG + j) * 2 + 0] * INV_2PI;  // fx / 2pi
                g12[4 + j] = gf[(idx * GG + j) * 2 + 1] * INV_2PI;  // fy / 2pi
                g12[8 + j] = gw[idx * GG + j];                      // weight
            }
        }
        float* gp = gab + (size_t)(i - NLpad) * 16;
        #pragma unroll
        for (int j = 0; j < 12; ++j) gp[j] = g12[j];
    }
}

// -------------------- kernel 2: raster (WMMA sigma) ------------------------
// Wave = one 16-pixel row segment. Chunk = 16 gaussians.
// D layout: lane l, vgpr v -> pixel M = v + 8*(l>>4), gaussian N = l&15.
__global__ __launch_bounds__(256) void raster_kernel(
    const float* __restrict__ base, const float* __restrict__ gab,
    float* __restrict__ out, int NLpad, int NHpad)
{
    const int lane = threadIdx.x & 31;
    const int wv   = threadIdx.x >> 5;
    const int tile = blockIdx.x * 8 + wv;           // 0..4095
    const int ty   = tile >> 4;                     // row
    const int tx   = (tile & 15) << 4;              // x base of 16-pixel segment
    const int half = lane >> 4;                     // 0 or 1
    const int col  = lane & 15;                     // gaussian column in chunk

    // A fragment (constant): lane holds M=col, K = 2*half + {0,1}
    // rows: [x^2, x, 1, 0] with tile-local x = col + 0.5
    const float xm = (float)col + 0.5f;
    v2f afrag;
    afrag.x = half ? 1.0f : xm * xm;
    afrag.y = half ? 0.0f : xm;

    const float cyb = (float)ty + 0.5f;

    float acr[8], acg[8], acb[8];
    #pragma unroll
    for (int v = 0; v < 8; ++v) { acr[v] = 0.0f; acg[v] = 0.0f; acb[v] = 0.0f; }

    // ---------------- low (pure gaussian) chunks ----------------
    const int lowChunks = NLpad >> 4;
    #pragma unroll 2
    for (int c = 0; c < lowChunks; ++c) {
        const float* p = base + (size_t)(c * 16 + col) * 16;
        const float cx = p[0], cy = p[1];
        const float A  = p[2], Bc = p[3], Cc = p[4];
        const float op = p[5];
        const float cr = p[6], cg = p[7], cb = p[8];

        const float cxp = cx - (float)tx;
        const float cyp = cy - cyb;
        // coefficients pre-scaled by -log2e so WMMA yields log2(gauss weight)
        const float As = NEG_LOG2E * A;
        const float e1 = NEG_LOG2E * (-(2.0f * A * cxp + Bc * cyp));
        const float e0 = NEG_LOG2E * ((A * cxp + Bc * cyp) * cxp + Cc * cyp * cyp);

        v2f bfrag;
        bfrag.x = half ? e0 : As;
        bfrag.y = half ? 0.0f : e1;

        v8f cz = {};
        v8f lg = __builtin_amdgcn_wmma_f32_16x16x4_f32(
            false, afrag, false, bfrag, (short)0, cz, false, false);

        #pragma unroll
        for (int v = 0; v < 8; ++v) {
            float w = op * __builtin_amdgcn_exp2f(lg[v]);   // raw v_exp_f32
            acr[v] = fmaf(w, cr, acr[v]);
            acg[v] = fmaf(w, cg, acg[v]);
            acb[v] = fmaf(w, cb, acb[v]);
        }
    }

    // ---------------- high (gabor-modulated) chunks ----------------
    const int highChunks = NHpad >> 4;
    const float* baseH = base + (size_t)NLpad * 16;
    for (int c = 0; c < highChunks; ++c) {
        const float* p = baseH + (size_t)(c * 16 + col) * 16;
        const float cx = p[0], cy = p[1];
        const float A  = p[2], Bc = p[3], Cc = p[4];
        const float op = p[5];
        const float cr = p[6], cg = p[7], cb = p[8];

        const float cxp = cx - (float)tx;
        const float cyp = cy - cyb;
        const float As = NEG_LOG2E * A;
        const float e1 = NEG_LOG2E * (-(2.0f * A * cxp + Bc * cyp));
        const float e0 = NEG_LOG2E * ((A * cxp + Bc * cyp) * cxp + Cc * cyp * cyp);

        v2f bfrag;
        bfrag.x = half ? e0 : As;
        bfrag.y = half ? 0.0f : e1;

        v8f cz = {};
        v8f lg = __builtin_amdgcn_wmma_f32_16x16x4_f32(
            false, afrag, false, bfrag, (short)0, cz, false, false);

        const float* q = gab + (size_t)(c * 16 + col) * 16;
        const float fx0 = q[0], fx1 = q[1], fx2 = q[2], fx3 = q[3];
        const float fy0 = q[4], fy1 = q[5], fy2 = q[6], fy3 = q[7];
        const float w0  = q[8], w1  = q[9], w2  = q[10], w3 = q[11];

        const float dy = -cyp;
        const float py0 = fy0 * dy, py1 = fy1 * dy, py2 = fy2 * dy, py3 = fy3 * dy;

        #pragma unroll
        for (int v = 0; v < 8; ++v) {
            float w = op * __builtin_amdgcn_exp2f(lg[v]);
            // pixel M = v + 8*half ; dx = (M + 0.5) - cxp (tile-local)
            float dxp = ((float)(v + 8 * half) + 0.5f) - cxp;
            // fx,fy already scaled by 1/2pi -> raw v_cos_f32
            float mod = w0 * __builtin_amdgcn_cosf(fmaf(fx0, dxp, py0))
                      + w1 * __builtin_amdgcn_cosf(fmaf(fx1, dxp, py1))
                      + w2 * __builtin_amdgcn_cosf(fmaf(fx2, dxp, py2))
                      + w3 * __builtin_amdgcn_cosf(fmaf(fx3, dxp, py3));
            w *= mod;
            acr[v] = fmaf(w, cr, acr[v]);
            acg[v] = fmaf(w, cg, acg[v]);
            acb[v] = fmaf(w, cb, acb[v]);
        }
    }

    // Butterfly-reduce partial sums over the 16 gaussian-columns of each half
    #pragma unroll
    for (int v = 0; v < 8; ++v) {
        #pragma unroll
        for (int s = 1; s < 16; s <<= 1) {
            acr[v] += __shfl_xor(acr[v], s, 32);
            acg[v] += __shfl_xor(acg[v], s, 32);
            acb[v] += __shfl_xor(acb[v], s, 32);
        }
    }

    // One lane per half writes its 8 pixels (planes: out[k][y][x], k-major)
    if (col == 0) {
        #pragma unroll
        for (int v = 0; v < 8; ++v) {
            int M = v + 8 * half;
            int o = ty * WW + tx + M;
            float r = fminf(fmaxf(acr[v], 0.0f), 1.0f);
            float g = fminf(fmaxf(acg[v], 0.0f), 1.0f);
            float b = fminf(fmaxf(acb[v], 0.0f), 1.0f);
            out[o]               = r;
            out[HH * WW + o]     = g;
            out[2 * HH * WW + o] = b;
        }
    }
}

// -------------------- host launch ------------------------------------------
extern "C" void kernel_launch(void* const* d_in, const int* in_sizes, int n_in,
                              void* d_out, int out_size, void* d_ws, size_t ws_size,
                              hipStream_t stream) {
    const float* low_mu    = (const float*)d_in[0];
    const float* high_mu   = (const float*)d_in[1];
    const float* low_chol  = (const float*)d_in[2];
    const float* high_chol = (const float*)d_in[3];
    const float* low_feat  = (const float*)d_in[4];
    const float* high_feat = (const float*)d_in[5];
    const float* low_opac  = (const float*)d_in[6];
    const float* high_opac = (const float*)d_in[7];
    const float* gfreqs    = (const float*)d_in[8];
    const float* gweights  = (const float*)d_in[9];

    int nl = in_sizes[0] / 2;
    int nh = in_sizes[1] / 2;
    int NLpad = (nl + 15) & ~15;
    int NHpad = (nh + 15) & ~15;

    float* base = (float*)d_ws;
    float* gab  = base + (size_t)(NLpad + NHpad) * 16;

    int tot = NLpad + NHpad;
    project_kernel<<<(tot + 255) / 256, 256, 0, stream>>>(
        low_mu, high_mu, low_chol, high_chol, low_feat, high_feat,
        low_opac, high_opac, gfreqs, gweights, base, gab, nl, nh, NLpad, NHpad);

    // 65536 pixels / 16 per wave / 8 waves per block = 512 blocks
    raster_kernel<<<(HH * WW / 16) / 8, 256, 0, stream>>>(
        base, gab, (float*)d_out, NLpad, NHpad);
}